// NashCascadeNeuralNetwork_60799557042186
// MI455X (gfx1250) — compile-verified
//
#include <hip/hip_runtime.h>

// Nash cascade network on gfx1250 (MI455X).
// Layers: BPL = [1, 1536, 1536, 1536, 1536]; NSPIG = [1536,1536,1536,1536,1]; T = 4.
// d_in: theta[L*MAX_B*MAX_S], sp_h[...], sp_a[...], H_init[L*MAX_B], precip[T]  (all f32)
// d_out: outflow[T] f32.

typedef __attribute__((ext_vector_type(2))) float v2f;
typedef __attribute__((ext_vector_type(8))) float v8f;

#define NB 1536
#define NS 1536
#define NLAYERS 5
#define G_CONST 9.81f
#define TWO_G 19.62f   // 2 * 9.81

// ---------------------------------------------------------------------------
// init: copy H_init into workspace state, zero the output vector.
// ---------------------------------------------------------------------------
__global__ void nash_init(const float* __restrict__ Hinit, float* __restrict__ Hws,
                          float* __restrict__ out, int outN) {
    int i = blockIdx.x * blockDim.x + threadIdx.x;
    if (i < NLAYERS * NB) Hws[i] = Hinit[i];
    if (i < outN) out[i] = 0.0f;
}

// ---------------------------------------------------------------------------
// Serial spigot recurrence, 4 steps, minimized dependent chain:
//   pre_s = 2G*(H0 - sp_h_s), c_s = theta_s*a_s   (off-chain)
//   hd = pre_s - g ; f = hd>0 ? c_s*sqrt(hd) : 0 ; g = fma(G, f, g)
// g tracks 2G*(0.5*cum); cum accumulates off the critical path.
// ---------------------------------------------------------------------------
__device__ __forceinline__ void spigot4(float H0, float& g, float& cum,
                                        float4 t4, float4 h4, float4 a4, float4& o) {
    float p0 = TWO_G * (H0 - h4.x), p1 = TWO_G * (H0 - h4.y);
    float p2 = TWO_G * (H0 - h4.z), p3 = TWO_G * (H0 - h4.w);
    float c0 = t4.x * a4.x, c1 = t4.y * a4.y, c2 = t4.z * a4.z, c3 = t4.w * a4.w;
    float hd;
    hd = p0 - g; o.x = (hd > 0.0f) ? c0 * sqrtf(hd) : 0.0f; g = fmaf(G_CONST, o.x, g);
    hd = p1 - g; o.y = (hd > 0.0f) ? c1 * sqrtf(hd) : 0.0f; g = fmaf(G_CONST, o.y, g);
    hd = p2 - g; o.z = (hd > 0.0f) ? c2 * sqrtf(hd) : 0.0f; g = fmaf(G_CONST, o.z, g);
    hd = p3 - g; o.w = (hd > 0.0f) ? c3 * sqrtf(hd) : 0.0f; g = fmaf(G_CONST, o.w, g);
    cum += (o.x + o.y) + (o.z + o.w);
}

// ---------------------------------------------------------------------------
// Layer 0: single bucket, 1536 spigots, sequential scan. Its s_q row IS the
// (pre-precip) inflow of layer 1, so write it straight into inflow[].
// ---------------------------------------------------------------------------
__global__ void nash_layer0(const float* __restrict__ theta, const float* __restrict__ sph,
                            const float* __restrict__ spa, float* __restrict__ H,
                            float* __restrict__ inflow, const float* __restrict__ precip, int t) {
    if (threadIdx.x != 0) return;
    float ppl = precip[t] / (float)NLAYERS;
    float H0 = H[0] + ppl;          // layer 0, bucket 0
    float g = 0.0f, cum = 0.0f;
    for (int s = 0; s < NS; s += 4) {
        float4 t4 = *(const float4*)(theta + s);
        float4 h4 = *(const float4*)(sph + s);
        float4 a4 = *(const float4*)(spa + s);
        float4 o;
        spigot4(H0, g, cum, t4, h4, a4, o);
        *(float4*)(inflow + s) = o;
    }
    H[0] = H0 - cum;
}

// ---------------------------------------------------------------------------
// Middle layers (1..3): one lane per bucket, sequential scan over 1536 spigots.
// Flows stored row-major sq[bucket][spigot] for the WMMA column reduction.
// 32-thread blocks: each serial wave gets its own WGP (48 waves total).
// ---------------------------------------------------------------------------
__global__ void nash_solve_layer(const float* __restrict__ theta, const float* __restrict__ sph,
                                 const float* __restrict__ spa, float* __restrict__ H,
                                 const float* __restrict__ inflow, const float* __restrict__ precip,
                                 int t, float* __restrict__ sq, int layer) {
    int b = blockIdx.x * blockDim.x + threadIdx.x;
    if (b >= NB) return;
    float ppl_term = precip[t] / (float)NLAYERS / (float)NB;
    float H0 = H[layer * NB + b] + inflow[b] + ppl_term;
    size_t base = ((size_t)layer * NB + b) * (size_t)NS;
    const float* th = theta + base;
    const float* sh = sph + base;
    const float* sa = spa + base;
    float* sqr = sq + (size_t)b * NS;
    float g = 0.0f, cum = 0.0f;
    for (int s = 0; s < NS; s += 4) {
        float4 t4 = *(const float4*)(th + s);
        float4 h4 = *(const float4*)(sh + s);
        float4 a4 = *(const float4*)(sa + s);
        float4 o;
        spigot4(H0, g, cum, t4, h4, a4, o);
        *(float4*)(sqr + s) = o;
    }
    H[layer * NB + b] = H0 - cum;
}

// ---------------------------------------------------------------------------
// Column sums via V_WMMA_F32_16X16X4_F32:  D = ones(16x4) x B(4x16) + C.
// With A == all-ones, every row of D equals the column sums of B over K.
// One wave per 16-column tile (96 waves total, 1536/16). EXEC is all-ones.
// Two accumulators break the WMMA->WMMA accumulation dependence.
// ---------------------------------------------------------------------------
__global__ void nash_colsum_wmma(const float* __restrict__ sq, float* __restrict__ inflow) {
    int wave  = blockIdx.x * (blockDim.x >> 5) + (threadIdx.x >> 5);
    int lane  = threadIdx.x & 31;
    int nbase = wave * 16;          // 96 waves exactly cover 1536 columns

    v2f ones; ones.x = 1.0f; ones.y = 1.0f;
    v8f acc0 = {};
    v8f acc1 = {};

    int col  = lane & 15;
    int rsel = (lane >> 4) << 1;    // lanes 0-15 -> rows {0,1}; lanes 16-31 -> rows {2,3}
    const float* p0 = sq + (size_t)(rsel + 0) * NS + nbase + col;  // feeds B VGPR0
    const float* p1 = sq + (size_t)(rsel + 1) * NS + nbase + col;  // feeds B VGPR1

    for (int k = 0; k < NB; k += 8) {
        v2f b0, b1;
        b0.x = p0[(size_t)k * NS];
        b0.y = p1[(size_t)k * NS];
        b1.x = p0[(size_t)(k + 4) * NS];
        b1.y = p1[(size_t)(k + 4) * NS];
        acc0 = __builtin_amdgcn_wmma_f32_16x16x4_f32(false, ones, false, b0,
                                                     (short)0, acc0, false, false);
        acc1 = __builtin_amdgcn_wmma_f32_16x16x4_f32(false, ones, false, b1,
                                                     (short)0, acc1, false, false);
    }
    // D VGPR0: lanes 0-15 hold row M=0 (N = lane) — all rows identical = column sums.
    float s = acc0[0] + acc1[0];
    if (lane < 16) inflow[nbase + lane] = s;
}

// ---------------------------------------------------------------------------
// Last layer (4): one spigot per bucket; network outflow = sum of flows.
// ---------------------------------------------------------------------------
__global__ void nash_layer_last(const float* __restrict__ theta, const float* __restrict__ sph,
                                const float* __restrict__ spa, float* __restrict__ H,
                                const float* __restrict__ inflow, const float* __restrict__ precip,
                                int t, float* __restrict__ out) {
    __shared__ float red[256];
    int b = blockIdx.x * blockDim.x + threadIdx.x;
    float flow = 0.0f;
    if (b < NB) {
        float ppl_term = precip[t] / (float)NLAYERS / (float)NB;
        float H0 = H[4 * NB + b] + inflow[b] + ppl_term;
        size_t base = ((size_t)4 * NB + b) * (size_t)NS;   // spigot 0
        float hd = H0 - sph[base];
        if (hd > 0.0f) flow = theta[base] * spa[base] * sqrtf(TWO_G * hd);
        H[4 * NB + b] = H0 - flow;
    }
    red[threadIdx.x] = flow;
    __syncthreads();
    for (int off = blockDim.x >> 1; off > 0; off >>= 1) {
        if ((int)threadIdx.x < off) red[threadIdx.x] += red[threadIdx.x + off];
        __syncthreads();
    }
    if (threadIdx.x == 0) atomicAdd(&out[t], red[0]);
}

// ---------------------------------------------------------------------------
extern "C" void kernel_launch(void* const* d_in, const int* in_sizes, int n_in,
                              void* d_out, int out_size, void* d_ws, size_t ws_size,
                              hipStream_t stream) {
    const float* theta  = (const float*)d_in[0];
    const float* sph    = (const float*)d_in[1];
    const float* spa    = (const float*)d_in[2];
    const float* Hinit  = (const float*)d_in[3];
    const float* precip = (const float*)d_in[4];
    float* out = (float*)d_out;
    float* ws  = (float*)d_ws;

    // workspace layout (floats): H[L*NB] | inflow[NB] | sq[NB*NS]  (~9.03 MB)
    float* H      = ws;
    float* inflow = ws + NLAYERS * NB;
    float* sq     = ws + NLAYERS * NB + NB;

    int T = in_sizes[4];

    nash_init<<<(NLAYERS * NB + 255) / 256, 256, 0, stream>>>(Hinit, H, out, out_size);

    for (int t = 0; t < T; ++t) {
        nash_layer0<<<1, 32, 0, stream>>>(theta, sph, spa, H, inflow, precip, t);
        for (int layer = 1; layer <= 3; ++layer) {
            // 48 blocks x 32 threads = one serial wave per WGP
            nash_solve_layer<<<NB / 32, 32, 0, stream>>>(
                theta, sph, spa, H, inflow, precip, t, sq, layer);
            // 12 blocks x 256 threads = 96 full waves, one per 16-column tile
            nash_colsum_wmma<<<12, 256, 0, stream>>>(sq, inflow);
        }
        nash_layer_last<<<(NB + 255) / 256, 256, 0, stream>>>(
            theta, sph, spa, H, inflow, precip, t, out);
    }
}